// RenderPipeline_48275432407664
// MI455X (gfx1250) — compile-verified
//
#include <hip/hip_runtime.h>
#include <math.h>

typedef float v2f __attribute__((ext_vector_type(2)));
typedef float v8f __attribute__((ext_vector_type(8)));

#define NPIX   4096      // 64*64
#define NTRI   7938
#define NCELL  3969      // 63*63
#define FLEN   360.04664611f   // (S-1)/(2*tan(radians(10)/2))
#define CCC    31.5f
#define EPS7   1e-7f

// One block per batch image. 1024 threads = 32 wave32 waves.
// LDS: 64KB z-buffer (128x128, uint-ordered positive floats) + WMMA staging.
__global__ __launch_bounds__(1024) void render_fused(
    const float* __restrict__ canon_depth,   // (16,1,64,64)
    const float* __restrict__ canon_img,     // (16,3,64,64)
    const float* __restrict__ views,         // (16,6)
    float* __restrict__ out,                 // [img 16*3*4096][depth 16*4096]
    float* __restrict__ uvz)                 // ws: 16*4096*3 floats
{
    __shared__ unsigned int zbuf[128 * 128];     // 65536 B
    __shared__ float stage[32][16][3];           // per-wave WMMA D extraction
    __shared__ float Mf[3][4];                   // forward affine [R | t]
    __shared__ float Mi[3][4];                   // inverse affine [R^T | t']

    const int b    = blockIdx.x;
    const int tid  = threadIdx.x;
    const int wave = tid >> 5;
    const int lane = tid & 31;
    const int hl   = lane & 15;

    // ---------------- Phase 0: per-batch matrices + zbuf init ----------------
    if (tid == 0) {
        const float* vw = views + b * 6;
        float rx = vw[0] * 1.04719755f;   // *60deg in rad
        float ry = vw[1] * 1.04719755f;
        float rz = vw[2] * 1.04719755f;
        float tx = vw[3] * 0.1f;
        float ty = vw[4] * 0.1f;
        float cx = cosf(rx), sx = sinf(rx);
        float cy = cosf(ry), sy = sinf(ry);
        float cz = cosf(rz), sz = sinf(rz);
        // R = Rz @ Ry @ Rx
        float R00 = cz * cy, R01 = cz * sy * sx - sz * cx, R02 = cz * sy * cx + sz * sx;
        float R10 = sz * cy, R11 = sz * sy * sx + cz * cx, R12 = sz * sy * cx - cz * sx;
        float R20 = -sy,     R21 = cy * sx,                R22 = cy * cx;
        // forward: p' = R*p + (center + T - R*center), center = (0,0,1)
        Mf[0][0] = R00; Mf[0][1] = R01; Mf[0][2] = R02; Mf[0][3] = tx - R02;
        Mf[1][0] = R10; Mf[1][1] = R11; Mf[1][2] = R12; Mf[1][3] = ty - R12;
        Mf[2][0] = R20; Mf[2][1] = R21; Mf[2][2] = R22; Mf[2][3] = 1.0f - R22;
        // inverse: q = R^T*p + (center - R^T*(T+center)), (T+center) = (tx,ty,1)
        Mi[0][0] = R00; Mi[0][1] = R10; Mi[0][2] = R20; Mi[0][3] = 0.0f - (R00 * tx + R10 * ty + R20);
        Mi[1][0] = R01; Mi[1][1] = R11; Mi[1][2] = R21; Mi[1][3] = 0.0f - (R01 * tx + R11 * ty + R21);
        Mi[2][0] = R02; Mi[2][1] = R12; Mi[2][2] = R22; Mi[2][3] = 1.0f - (R02 * tx + R12 * ty + R22);
    }
    for (int i = tid; i < 128 * 128; i += 1024) zbuf[i] = 0x41200000u;  // bits(10.0f) = FAR
    __syncthreads();

    // ---------------- Phase 1: vertex transform via WMMA f32 16x16x4 ----------------
    // D[m][n] = sum_k A[m][k]*B[k][n], A row m = (bx*d, by*d, d, 1), B[k][n] = Mf[n][k].
    float* my_uvz = uvz + (size_t)b * NPIX * 3;
    for (int it = 0; it < 8; ++it) {
        int p0 = (it * 32 + wave) * 16;
        int p  = p0 + hl;
        float dpt = canon_depth[b * NPIX + p];
        v2f a;
        if (lane < 16) {           // K = 0,1
            int row = p >> 6, col = p & 63;
            a.x = (((float)col - CCC) * (1.0f / FLEN)) * dpt;
            a.y = (((float)row - CCC) * (1.0f / FLEN)) * dpt;
        } else {                   // K = 2,3
            a.x = dpt; a.y = 1.0f;
        }
        v2f bm;
        if (hl < 3) {
            bm.x = (lane < 16) ? Mf[hl][0] : Mf[hl][2];
            bm.y = (lane < 16) ? Mf[hl][1] : Mf[hl][3];
        } else { bm.x = 0.0f; bm.y = 0.0f; }
        v8f c0 = {0.f, 0.f, 0.f, 0.f, 0.f, 0.f, 0.f, 0.f};
        v8f dm = __builtin_amdgcn_wmma_f32_16x16x4_f32(
            false, a, false, bm, (short)0, c0, false, false);
        if (hl < 3) {              // lanes holding components n = 0..2
            int base = (lane < 16) ? 0 : 8;
            #pragma unroll
            for (int r = 0; r < 8; ++r) stage[wave][base + r][hl] = dm[r];
        }
        __syncthreads();
        if (lane < 16) {
            float X = stage[wave][hl][0];
            float Y = stage[wave][hl][1];
            float Z = stage[wave][hl][2];
            float zs  = (Z > EPS7) ? Z : 1.0f;
            float inv = 1.0f / zs;
            float* o = my_uvz + p * 3;
            o[0] = X * inv * FLEN + CCC;   // screen u
            o[1] = Y * inv * FLEN + CCC;   // screen v
            o[2] = Z;
        }
        __syncthreads();
    }

    // ---------------- Phase 2: scatter rasterization into LDS z-buffer ----------------
    for (int t = tid; t < NTRI; t += 1024) {
        int cell = (t < NCELL) ? t : (t - NCELL);
        int rr = cell / 63, q0 = cell - rr * 63;
        int v00 = (rr << 6) + q0;
        int i0, i1, i2;
        if (t < NCELL) { i0 = v00;     i1 = v00 + 1;  i2 = v00 + 64; }
        else           { i0 = v00 + 1; i1 = v00 + 65; i2 = v00 + 64; }
        const float* A3 = my_uvz + i0 * 3;
        const float* B3 = my_uvz + i1 * 3;
        const float* C3 = my_uvz + i2 * 3;
        float ax = A3[0], ay = A3[1], az = A3[2];
        float bx = B3[0], by = B3[1], bz = B3[2];
        float gx = C3[0], gy = C3[1], gz = C3[2];
        if (!(az > EPS7 && bz > EPS7 && gz > EPS7)) continue;   // invalid -> FAR (no-op)
        float ra = 1.0f / az, rb = 1.0f / bz, rc = 1.0f / gz;
        float umin = fminf(ax, fminf(bx, gx)), umax = fmaxf(ax, fmaxf(bx, gx));
        float vmin = fminf(ay, fminf(by, gy)), vmax = fmaxf(ay, fmaxf(by, gy));
        // pixel centers: px = 0.5*x - 0.25  ->  x = 2*px + 0.5
        int xlo = (int)fminf(fmaxf(ceilf (2.0f * umin + 0.5f),  0.0f), 128.0f);
        int xhi = (int)fmaxf(fminf(floorf(2.0f * umax + 0.5f), 127.0f), -1.0f);
        int ylo = (int)fminf(fmaxf(ceilf (2.0f * vmin + 0.5f),  0.0f), 128.0f);
        int yhi = (int)fmaxf(fminf(floorf(2.0f * vmax + 0.5f), 127.0f), -1.0f);
        for (int y = ylo; y <= yhi; ++y) {
            float py = 0.5f * (float)y - 0.25f;
            for (int x = xlo; x <= xhi; ++x) {
                float px = 0.5f * (float)x - 0.25f;
                float w0 = (gx - bx) * (py - by) - (gy - by) * (px - bx);
                float w1 = (ax - gx) * (py - gy) - (ay - gy) * (px - gx);
                float w2 = (bx - ax) * (py - ay) - (by - ay) * (px - ax);
                float area = w0 + w1 + w2;
                bool inside = (w0 >= 0.f && w1 >= 0.f && w2 >= 0.f) ||
                              (w0 <= 0.f && w1 <= 0.f && w2 <= 0.f);
                if (!inside || fabsf(area) <= 1e-10f) continue;
                float invz = (w0 * ra + w1 * rb + w2 * rc) / area;
                if (invz <= 1e-10f) continue;
                float depth = 1.0f / invz;
                if (depth > 0.1f && depth < 10.0f)
                    atomicMin(&zbuf[(y << 7) + x], __float_as_uint(depth));
            }
        }
    }
    __syncthreads();

    // ---------------- Phase 3: 2x2 AA mean + clip; reuse zbuf[0..4095] as depth ----------------
    float dreg[4];
    float* out_depth = out + (size_t)gridDim.x * 3 * NPIX;
    #pragma unroll
    for (int k = 0; k < 4; ++k) {
        int idx = tid + k * 1024;
        int y = idx >> 6, x = idx & 63;
        int o = (y << 8) + (x << 1);
        float s = __uint_as_float(zbuf[o])       + __uint_as_float(zbuf[o + 1]) +
                  __uint_as_float(zbuf[o + 128]) + __uint_as_float(zbuf[o + 129]);
        float dmean = 0.25f * s;
        dmean = fminf(fmaxf(dmean, 0.8f), 1.2f);   // clip(MIN_D-margin, MAX_D+margin)
        dreg[k] = dmean;
        out_depth[(size_t)b * NPIX + idx] = dmean;
    }
    __syncthreads();
    #pragma unroll
    for (int k = 0; k < 4; ++k) zbuf[tid + k * 1024] = __float_as_uint(dreg[k]);
    __syncthreads();

    // ---------------- Phase 4: inverse warp via WMMA + bilinear grid-sample ----------------
    const float* img = canon_img + (size_t)b * 3 * NPIX;
    float* out_img = out + (size_t)b * 3 * NPIX;
    for (int it = 0; it < 8; ++it) {
        int p0 = (it * 32 + wave) * 16;
        int p  = p0 + hl;
        float dd = __uint_as_float(zbuf[p]);
        v2f a;
        if (lane < 16) {
            int row = p >> 6, col = p & 63;
            a.x = (((float)col - CCC) * (1.0f / FLEN)) * dd;
            a.y = (((float)row - CCC) * (1.0f / FLEN)) * dd;
        } else {
            a.x = dd; a.y = 1.0f;
        }
        v2f bm;
        if (hl < 3) {
            bm.x = (lane < 16) ? Mi[hl][0] : Mi[hl][2];
            bm.y = (lane < 16) ? Mi[hl][1] : Mi[hl][3];
        } else { bm.x = 0.0f; bm.y = 0.0f; }
        v8f c0 = {0.f, 0.f, 0.f, 0.f, 0.f, 0.f, 0.f, 0.f};
        v8f dm = __builtin_amdgcn_wmma_f32_16x16x4_f32(
            false, a, false, bm, (short)0, c0, false, false);
        if (hl < 3) {
            int base = (lane < 16) ? 0 : 8;
            #pragma unroll
            for (int r = 0; r < 8; ++r) stage[wave][base + r][hl] = dm[r];
        }
        __syncthreads();
        if (lane < 16) {
            float qx = stage[wave][hl][0];
            float qy = stage[wave][hl][1];
            float qz = stage[wave][hl][2];
            float inv = 1.0f / (qz + EPS7);
            float u = (FLEN * qx + CCC * qz) * inv;   // K @ (pc/(z+eps)), row 0
            float v = (FLEN * qy + CCC * qz) * inv;
            float x0f = floorf(u), y0f = floorf(v);
            float wx = u - x0f, wy = v - y0f;
            float r0 = 0.f, r1 = 0.f, r2 = 0.f;
            #pragma unroll
            for (int tp = 0; tp < 4; ++tp) {
                float xf = x0f + (float)(tp & 1);
                float yf = y0f + (float)(tp >> 1);
                float w = ((tp & 1) ? wx : 1.0f - wx) * ((tp >> 1) ? wy : 1.0f - wy);
                bool valid = (xf >= 0.0f) && (xf <= 63.0f) && (yf >= 0.0f) && (yf <= 63.0f);
                if (!valid) w = 0.0f;
                int xi = (int)fminf(fmaxf(xf, 0.0f), 63.0f);
                int yi = (int)fminf(fmaxf(yf, 0.0f), 63.0f);
                const float* tpp = img + (yi << 6) + xi;
                r0 += w * tpp[0];
                r1 += w * tpp[NPIX];
                r2 += w * tpp[2 * NPIX];
            }
            out_img[p]            = r0;
            out_img[NPIX + p]     = r1;
            out_img[2 * NPIX + p] = r2;
        }
        __syncthreads();
    }
}

extern "C" void kernel_launch(void* const* d_in, const int* in_sizes, int n_in,
                              void* d_out, int out_size, void* d_ws, size_t ws_size,
                              hipStream_t stream) {
    const float* canon_depth = (const float*)d_in[0];
    const float* canon_img   = (const float*)d_in[1];
    const float* views       = (const float*)d_in[2];
    float* out = (float*)d_out;
    float* uvz = (float*)d_ws;   // 16*4096*3 floats = 768 KB scratch
    (void)in_sizes; (void)n_in; (void)out_size; (void)ws_size;
    render_fused<<<dim3(16), dim3(1024), 0, stream>>>(canon_depth, canon_img, views, out, uvz);
}